// CapsuleLayer_1288490189211
// MI455X (gfx1250) — compile-verified
//
#include <hip/hip_runtime.h>
#include <hip/hip_bf16.h>
#include <math.h>

// ---------------- problem constants ----------------
#define B_DIM   8
#define S_DIM   2048
#define I_DIM   512      // GEMM K
#define N_CAPS  32
#define D_DIM   64
#define ND      2048     // N_CAPS * D_DIM (GEMM N)

// ---------------- GEMM tiling ----------------
#define MB 128           // block M tile (s)
#define NB 64            // block N tile (= one capsule n, contiguous W rows)
#define KB 16            // K chunk staged through LDS
#define LDA 20           // padded LDS stride (floats): 16B-aligned rows, conflict-free
#define LDB 20

typedef float v2f __attribute__((ext_vector_type(2)));
typedef float v8f __attribute__((ext_vector_type(8)));

// capsules[b][s][n*64+d] = sum_i x[b][s][i] * W[n][i][d]
// One workgroup: 128(M) x 64(N) tile, 8 waves in 4x2 layout, each wave 32x32
// (2x2 wmma accumulators). Uses V_WMMA_F32_16X16X4_F32 (exact fp32 on matrix pipe).
__global__ __launch_bounds__(256) void caps_gemm(const float* __restrict__ x,
                                                 const float* __restrict__ W,
                                                 float* __restrict__ cap)
{
    __shared__ __align__(16) float As[MB * LDA];   // [m][k], stride 20
    __shared__ __align__(16) float Bs[NB * LDB];   // [c][k] (transposed), stride 20

    const int b    = blockIdx.z;
    const int s0   = blockIdx.x * MB;
    const int n    = blockIdx.y;            // nd tile = [n*64, n*64+64)
    const int t    = threadIdx.x;
    const int lane = t & 31;
    const int wave = t >> 5;
    const int wm   = wave & 3;              // M sub-tile (32 rows each)
    const int wn   = wave >> 2;             // N sub-tile (32 cols each)
    const int lhalf = lane >> 4;            // 0: K=k0,k0+1 ; 1: K=k0+2,k0+3
    const int l16   = lane & 15;

    v8f acc[2][2] = {};

    const float* xg = x + ((size_t)b * S_DIM + s0) * I_DIM;
    const float* wg = W + (size_t)n * I_DIM * D_DIM;

    for (int k0 = 0; k0 < I_DIM; k0 += KB) {
        __syncthreads();   // protect LDS from previous chunk's readers
        // ---- stage A: 128x16 floats (2 float4 per thread) ----
        {
            int idx = t;
#pragma unroll
            for (int r = 0; r < 2; ++r, idx += 256) {
                int m  = idx >> 2;
                int kq = (idx & 3) << 2;
                float4 v = *(const float4*)(xg + (size_t)m * I_DIM + k0 + kq);
                *(float4*)(&As[m * LDA + kq]) = v;
            }
        }
        // ---- stage B: 64(k) x 16(d) floats, transposed into Bs[c][k] ----
        {
            int kk = t >> 2;
            int d4 = (t & 3) << 2;
            float4 v = *(const float4*)(wg + (size_t)(k0 + kk) * D_DIM + d4);
            Bs[(d4 + 0) * LDB + kk] = v.x;
            Bs[(d4 + 1) * LDB + kk] = v.y;
            Bs[(d4 + 2) * LDB + kk] = v.z;
            Bs[(d4 + 3) * LDB + kk] = v.w;
        }
        __syncthreads();
        // ---- compute: 4 K-steps of 4 wmma ----
#pragma unroll
        for (int k4 = 0; k4 < KB; k4 += 4) {
            const int kA = k4 + lhalf * 2;
            v2f a0 = *(const v2f*)(&As[(wm * 32 +  0 + l16) * LDA + kA]);
            v2f a1 = *(const v2f*)(&As[(wm * 32 + 16 + l16) * LDA + kA]);
            v2f b0 = *(const v2f*)(&Bs[(wn * 32 +  0 + l16) * LDB + kA]);
            v2f b1 = *(const v2f*)(&Bs[(wn * 32 + 16 + l16) * LDB + kA]);
            acc[0][0] = __builtin_amdgcn_wmma_f32_16x16x4_f32(false, a0, false, b0,
                          (short)0, acc[0][0], false, false);
            acc[0][1] = __builtin_amdgcn_wmma_f32_16x16x4_f32(false, a0, false, b1,
                          (short)0, acc[0][1], false, false);
            acc[1][0] = __builtin_amdgcn_wmma_f32_16x16x4_f32(false, a1, false, b0,
                          (short)0, acc[1][0], false, false);
            acc[1][1] = __builtin_amdgcn_wmma_f32_16x16x4_f32(false, a1, false, b1,
                          (short)0, acc[1][1], false, false);
        }
    }

    // ---- store: C/D layout — VGPR v: lanes0-15 M=v, lanes16-31 M=v+8 ----
    float* cg = cap + ((size_t)b * S_DIM + s0) * ND + (size_t)n * D_DIM;
#pragma unroll
    for (int sm = 0; sm < 2; ++sm)
#pragma unroll
        for (int sn = 0; sn < 2; ++sn) {
            const int col = wn * 32 + sn * 16 + l16;
#pragma unroll
            for (int v = 0; v < 8; ++v) {
                const int row = wm * 32 + sm * 16 + v + lhalf * 8;
                cg[(size_t)row * ND + col] = acc[sm][sn][v];
            }
        }
}

// ---------------- routing ----------------
#define SCH     64     // s rows per chunk
#define NCHUNK  32     // S_DIM / SCH

// partial[b][chunk][nd] = sum_{s in chunk} softmax_n(blog[b,s,:])[n] * cap[b,s,nd]
__global__ __launch_bounds__(256) void nodes_partial(const float* __restrict__ cap,
                                                     const float* __restrict__ blog,
                                                     float* __restrict__ partial,
                                                     int iter0)
{
    __shared__ float rS[SCH * N_CAPS];
    const int b     = blockIdx.y;
    const int chunk = blockIdx.x;
    const int s0    = chunk * SCH;
    const int t     = threadIdx.x;

    if (t < SCH) {
        if (iter0) {
#pragma unroll
            for (int n = 0; n < N_CAPS; ++n) rS[t * N_CAPS + n] = 1.0f / 32.0f;
        } else {
            const float* bl = blog + ((size_t)b * S_DIM + s0 + t) * N_CAPS;
            float v[N_CAPS];
            float mx = -3.4e38f;
#pragma unroll
            for (int n = 0; n < N_CAPS; ++n) { v[n] = bl[n]; mx = fmaxf(mx, v[n]); }
            float sum = 0.f;
#pragma unroll
            for (int n = 0; n < N_CAPS; ++n) { v[n] = __expf(v[n] - mx); sum += v[n]; }
            const float inv = 1.0f / sum;
#pragma unroll
            for (int n = 0; n < N_CAPS; ++n) rS[t * N_CAPS + n] = v[n] * inv;
        }
    }
    __syncthreads();

    const int nd0 = t * 4;                 // + second strip at +1024
    const int n0  = nd0 >> 6;
    const int n1  = (nd0 + 1024) >> 6;
    float4 acc0 = {0, 0, 0, 0}, acc1 = {0, 0, 0, 0};
    const float* cg = cap + ((size_t)b * S_DIM + s0) * ND;
    for (int s = 0; s < SCH; ++s) {
        const float* row = cg + (size_t)s * ND;
        const float r0 = rS[s * N_CAPS + n0];
        const float r1 = rS[s * N_CAPS + n1];
        const float4 c0 = *(const float4*)(row + nd0);
        const float4 c1 = *(const float4*)(row + nd0 + 1024);
        acc0.x += r0 * c0.x; acc0.y += r0 * c0.y; acc0.z += r0 * c0.z; acc0.w += r0 * c0.w;
        acc1.x += r1 * c1.x; acc1.y += r1 * c1.y; acc1.z += r1 * c1.z; acc1.w += r1 * c1.w;
    }
    float* pg = partial + ((size_t)(b * NCHUNK + chunk)) * ND;
    *(float4*)(pg + nd0)        = acc0;
    *(float4*)(pg + nd0 + 1024) = acc1;
}

// outp[b*2048+nd] = (tanh of) sum_chunk partial[b][chunk][nd]
__global__ void reduce_nodes(const float* __restrict__ partial,
                             float* __restrict__ outp, int apply_tanh)
{
    const int idx = blockIdx.x * blockDim.x + threadIdx.x;   // 0..16383
    const int b  = idx >> 11;
    const int nd = idx & (ND - 1);
    float s = 0.f;
#pragma unroll
    for (int c = 0; c < NCHUNK; ++c)
        s += partial[((size_t)(b * NCHUNK + c)) * ND + nd];
    outp[idx] = apply_tanh ? tanhf(s) : s;
}

// blog[b,s,n] (+)= sum_d cap[b,s,n,d] * tn[b,n,d]
__global__ __launch_bounds__(256) void agree_update(const float* __restrict__ cap,
                                                    const float* __restrict__ tn,
                                                    float* __restrict__ blog,
                                                    int first)
{
    __shared__ float tS[ND];
    const int b = blockIdx.y;
    const int t = threadIdx.x;
#pragma unroll
    for (int k = 0; k < 8; ++k) tS[t + k * 256] = tn[(size_t)b * ND + t + k * 256];
    __syncthreads();

    const int n = t & 31;
    const int s = blockIdx.x * 8 + (t >> 5);
    const float* row = cap + ((size_t)b * S_DIM + s) * ND + n * D_DIM;
    float sum = 0.f;
#pragma unroll
    for (int j = 0; j < D_DIM; j += 4) {
        const float4 c = *(const float4*)(row + j);
        sum += c.x * tS[n * 64 + j]     + c.y * tS[n * 64 + j + 1]
             + c.z * tS[n * 64 + j + 2] + c.w * tS[n * 64 + j + 3];
    }
    const size_t bi = ((size_t)b * S_DIM + s) * N_CAPS + n;
    blog[bi] = first ? sum : (blog[bi] + sum);
}

// ---------------- host orchestration ----------------
extern "C" void kernel_launch(void* const* d_in, const int* in_sizes, int n_in,
                              void* d_out, int out_size, void* d_ws, size_t ws_size,
                              hipStream_t stream)
{
    const float* x = (const float*)d_in[0];   // [8,2048,512]
    const float* W = (const float*)d_in[1];   // [32,512,64]
    float* out = (float*)d_out;               // [8,32,64]

    // workspace layout (~131 MB); capsules stays L2-resident (192 MB L2)
    const size_t CAP_BYTES  = (size_t)B_DIM * S_DIM * ND * sizeof(float);          // 128 MB
    const size_t PART_BYTES = (size_t)B_DIM * NCHUNK * ND * sizeof(float);         // 2 MB
    const size_t BLOG_BYTES = (size_t)B_DIM * S_DIM * N_CAPS * sizeof(float);      // 2 MB
    char* ws = (char*)d_ws;
    float* cap     = (float*)ws;
    float* partial = (float*)(ws + CAP_BYTES);
    float* blog    = (float*)(ws + CAP_BYTES + PART_BYTES);
    float* tn      = (float*)(ws + CAP_BYTES + PART_BYTES + BLOG_BYTES);           // 64 KB

    // 1) capsules GEMM (fp32 WMMA)
    dim3 gG(S_DIM / MB, ND / NB, B_DIM);      // 16 x 32 x 8
    caps_gemm<<<gG, 256, 0, stream>>>(x, W, cap);

    dim3 gN(NCHUNK, B_DIM);                   // 32 x 8
    dim3 gA(S_DIM / 8, B_DIM);                // 256 x 8

    // iteration 0: r = 1/32 (softmax of zeros); blog is overwritten (no init needed)
    nodes_partial<<<gN, 256, 0, stream>>>(cap, blog, partial, 1);
    reduce_nodes<<<64, 256, 0, stream>>>(partial, tn, 1);
    agree_update<<<gA, 256, 0, stream>>>(cap, tn, blog, 1);
    // iteration 1
    nodes_partial<<<gN, 256, 0, stream>>>(cap, blog, partial, 0);
    reduce_nodes<<<64, 256, 0, stream>>>(partial, tn, 1);
    agree_update<<<gA, 256, 0, stream>>>(cap, tn, blog, 0);
    // iteration 2: final nodes (no tanh) straight to d_out
    nodes_partial<<<gN, 256, 0, stream>>>(cap, blog, partial, 0);
    reduce_nodes<<<64, 256, 0, stream>>>(partial, out, 0);

    (void)in_sizes; (void)n_in; (void)out_size; (void)ws_size;
}